// SVD_22986664968525
// MI455X (gfx1250) — compile-verified
//
#include <hip/hip_runtime.h>

typedef __attribute__((ext_vector_type(2))) float v2f;
typedef __attribute__((ext_vector_type(8))) float v8f;

#define WAVES_PER_BLOCK 4

// One wave32 computes 16 samples:
//   D(16x16) = A(16x64:p rows) x B(64x16:q cols) via 16 chained f32 WMMAs.
//   diag(D)[s] = dot(p_s, q_s).  Features restaged via LDS -> coalesced b128 stores.
__global__ __launch_bounds__(128) void svd_wmma_kernel(
    const int*   __restrict__ user_item,
    const float* __restrict__ pu,
    const float* __restrict__ qi,
    const float* __restrict__ bu,
    const float* __restrict__ bi,
    const float* __restrict__ gmean,
    float*       __restrict__ out_pred,
    float*       __restrict__ out_feat,
    int batch)
{
    __shared__ float lds_feat[WAVES_PER_BLOCK][16 * 128];
    __shared__ float lds_pred[WAVES_PER_BLOCK][16];

    const int lane = threadIdx.x & 31;
    const int wv   = threadIdx.x >> 5;
    const int ntiles = batch >> 4;
    int tile = blockIdx.x * WAVES_PER_BLOCK + wv;
    if (tile >= ntiles) tile = ntiles - 1;   // safety clamp (dup work, same result)

    const int b0 = tile * 16;
    const int s  = lane & 15;   // sample within tile (row of A / col of B)
    const int hi = lane >> 4;   // which K-pair of the 4-wide K block this lane owns

    // per-lane sample ids (lanes L and L+16 load the same sample's ids)
    const int2 ui  = ((const int2*)user_item)[b0 + s];
    const int  uid = ui.x;
    const int  iid = ui.y;

    const float* prow = pu + (size_t)uid * 64 + hi * 2;
    const float* qrow = qi + (size_t)iid * 64 + hi * 2;

    // A/B fragments: lane holds elements {4*kb + 2*hi, 4*kb + 2*hi + 1} of its row
    v2f ap[16], bq[16];
#pragma unroll
    for (int kb = 0; kb < 16; ++kb) {
        ap[kb] = *(const v2f*)(prow + 4 * kb);
        bq[kb] = *(const v2f*)(qrow + 4 * kb);
    }

    // chained f32 WMMA: D = sum_kb A_kb * B_kb  (EXEC all-1s here)
    v8f acc = {};
#pragma unroll
    for (int kb = 0; kb < 16; ++kb) {
        acc = __builtin_amdgcn_wmma_f32_16x16x4_f32(
            /*neg_a=*/false, ap[kb],
            /*neg_b=*/false, bq[kb],
            /*c_mod=*/(short)0, acc,
            /*reuse_a=*/false, /*reuse_b=*/false);
    }

    // stage feature tile [16 rows x 128 floats] in LDS: row s = [p_s | q_s]
    float* slice = &lds_feat[wv][0];
#pragma unroll
    for (int kb = 0; kb < 16; ++kb) {
        *(v2f*)(slice + s * 128 +      4 * kb + 2 * hi) = ap[kb];
        *(v2f*)(slice + s * 128 + 64 + 4 * kb + 2 * hi) = bq[kb];
    }

    // diagonal of D: element (m,n) lives at lane n%16 (+16 bank), VGPR v with m = v + 8*hi
#pragma unroll
    for (int v = 0; v < 8; ++v) {
        if (v + 8 * hi == s) lds_pred[wv][s] = acc[v];
    }

    __syncthreads();

    // coalesced feature store: one 512B row per wave store (32 lanes x b128)
#pragma unroll
    for (int r = 0; r < 16; ++r) {
        const float4 f = *(const float4*)(slice + r * 128 + lane * 4);
        *(float4*)(out_feat + (size_t)(b0 + r) * 128 + lane * 4) = f;
    }

    // predict: lanes 0..15 own one sample each (ids already in registers)
    if (lane < 16) {
        float pr = gmean[0] + bu[uid] + bi[iid] + lds_pred[wv][s];
        pr = fminf(5.0f, fmaxf(1.0f, pr));
        out_pred[b0 + s] = pr;
    }
}

extern "C" void kernel_launch(void* const* d_in, const int* in_sizes, int n_in,
                              void* d_out, int out_size, void* d_ws, size_t ws_size,
                              hipStream_t stream) {
    const int*   user_item = (const int*)d_in[0];
    const float* pu        = (const float*)d_in[1];
    const float* qi        = (const float*)d_in[2];
    const float* bu        = (const float*)d_in[3];
    const float* bi        = (const float*)d_in[4];
    const float* gmean     = (const float*)d_in[5];

    const int batch = in_sizes[0] / 2;          // user_item is (BATCH, 2)
    float* out_pred = (float*)d_out;            // [batch]
    float* out_feat = (float*)d_out + batch;    // [batch, 128]

    const int ntiles = batch / 16;              // 8192 for BATCH=131072
    const int blocks = (ntiles + WAVES_PER_BLOCK - 1) / WAVES_PER_BLOCK;

    svd_wmma_kernel<<<blocks, WAVES_PER_BLOCK * 32, 0, stream>>>(
        user_item, pu, qi, bu, bi, gmean, out_pred, out_feat, batch);
}